// CustomClassificationLoss_48765058678812
// MI455X (gfx1250) — compile-verified
//
#include <hip/hip_runtime.h>

#define NUM_CLASSES 360
#define HALF 180
#define K1_BLOCKS 512
#define K1_THREADS 256
#define NWAVES (K1_BLOCKS * (K1_THREADS / 32)) /* 4096, multiple of 64 */

typedef float v2f __attribute__((ext_vector_type(2)));
typedef float v8f __attribute__((ext_vector_type(8)));

// ---------------------------------------------------------------------------
// Kernel 1: one wave per sample (16 samples/wave). Lane l covers circular
// distances r = 1 + l + 32k, k = 0..5.  Loads of row[(t +/- r) % 360] are
// contiguous across lanes -> coalesced.  Weights 0.98^r are per-(lane,k)
// registers, independent of the sample.  Per-lane accumulators are reduced
// across the wave with wave32 shuffles; one partial per wave goes to d_ws.
// ---------------------------------------------------------------------------
__global__ __launch_bounds__(K1_THREADS)
void ccl_partial_kernel(const float* __restrict__ logits,
                        const int* __restrict__ targets,
                        float* __restrict__ partials,
                        int B, int samples_per_wave) {
  const int gid   = blockIdx.x * K1_THREADS + threadIdx.x;
  const int lane  = threadIdx.x & 31;
  const int gwave = __builtin_amdgcn_readfirstlane(gid >> 5); // wave-uniform

  // 0.98^(1+lane+32k) for k = 0..5 (covers r = 1..179; k=5 partially active)
  float wk[6];
#pragma unroll
  for (int k = 0; k < 6; ++k)
    wk[k] = __powf(0.98f, (float)(1 + lane + 32 * k));

  int s_begin = gwave * samples_per_wave;
  int s_end   = s_begin + samples_per_wave;
  if (s_end > B) s_end = B;

  float acc = 0.0f;
  for (int s = s_begin; s < s_end; ++s) {
    const float* __restrict__ row = logits + (size_t)s * NUM_CLASSES;
    const int t = __builtin_amdgcn_readfirstlane(targets[s]); // wave-uniform

    if (s + 1 < s_end)
      __builtin_prefetch(row + NUM_CLASSES, 0, 0); // next sample's row

    if (lane == 0)
      acc += 1.0f - row[t]; // center term

#pragma unroll
    for (int k = 0; k < 6; ++k) {
      const int r = 1 + lane + 32 * k;
      if (r < HALF) { // k<5: always true (branch folds); k=5: lanes 0..18
        int ip = t + r; if (ip >= NUM_CLASSES) ip -= NUM_CLASSES;
        int im = t - r; if (im < 0)            im += NUM_CLASSES;
        const float gp = row[ip];
        const float gn = row[im];
        const float w  = wk[k];
        acc += fabsf(w - gp) + fabsf(w - gn);
      }
    }
  }

  // wave32 tree reduction
#pragma unroll
  for (int off = 16; off > 0; off >>= 1)
    acc += __shfl_down(acc, off, 32);

  if (lane == 0)
    partials[gwave] = acc; // every slot written every call (deterministic)
}

// ---------------------------------------------------------------------------
// Kernel 2: single wave. Reduce NWAVES partials with chained
// V_WMMA_F32_16X16X4_F32:  D = A(16x4 f32 tile of partials) x B(ones) + C.
// Each WMMA contracts 64 f32 values into the f32 accumulator (exact f32).
// A-layout (32-bit 16x4): lane L holds (M = L%16, K = 2*(L/16)) and K+1,
// so lane L loads the float2 at element M*4 + K  ->  off = (L&15)*4+(L>>4)*2.
// After the loop, column 0 lives in lanes 0 (rows 0-7) and 16 (rows 8-15).
// ---------------------------------------------------------------------------
__global__ __launch_bounds__(32)
void ccl_reduce_kernel(const float* __restrict__ partials, int n,
                       float* __restrict__ out, float invB) {
  const int lane = threadIdx.x;
  const int off  = (lane & 15) * 4 + (lane >> 4) * 2;

  v8f c = {0.f, 0.f, 0.f, 0.f, 0.f, 0.f, 0.f, 0.f};
  v2f bones = {1.0f, 1.0f};

  for (int base = 0; base < n; base += 64) {
    v2f a = *(const v2f*)(partials + base + off); // 8B-aligned, coalesced
    // 8 args: (neg_a, A, neg_b, B, c_mod, C, reuse_a, reuse_b)
    c = __builtin_amdgcn_wmma_f32_16x16x4_f32(false, a, false, bones,
                                              (short)0, c, false, false);
  }

  // lane 0: rows 0..7 of column 0; lane 16: rows 8..15 of column 0
  float colsum = c[0] + c[1] + c[2] + c[3] + c[4] + c[5] + c[6] + c[7];
  colsum += __shfl_down(colsum, 16, 32);

  if (lane == 0) {
    const float loss = colsum * invB;
    out[0] = loss;  // loss
    out[1] = 0.0f;  // constant 0 from reference tuple
    out[2] = loss;  // loss again
  }
}

// ---------------------------------------------------------------------------
extern "C" void kernel_launch(void* const* d_in, const int* in_sizes, int n_in,
                              void* d_out, int out_size, void* d_ws, size_t ws_size,
                              hipStream_t stream) {
  const float* logits  = (const float*)d_in[0];
  const int*   targets = (const int*)d_in[1];
  const int B = in_sizes[1]; // 65536

  float* ws = (float*)d_ws; // NWAVES floats = 16 KB of scratch

  const int spw = (B + NWAVES - 1) / NWAVES; // 16

  ccl_partial_kernel<<<K1_BLOCKS, K1_THREADS, 0, stream>>>(
      logits, targets, ws, B, spw);

  ccl_reduce_kernel<<<1, 32, 0, stream>>>(
      ws, NWAVES, (float*)d_out, 1.0f / (float)B);
}